// Memory_24438363915056
// MI455X (gfx1250) — compile-verified
//
#include <hip/hip_runtime.h>
#include <hip/hip_bf16.h>

// Problem constants (match reference)
#define NDIM 8192
#define GAMMA_F 0.95f

typedef __attribute__((ext_vector_type(2))) float v2f;
typedef __attribute__((ext_vector_type(8))) float v8f;

// ---------------------------------------------------------------------------
// Block-wide reductions (deterministic tree order)
// ---------------------------------------------------------------------------
__device__ __forceinline__ float block_reduce_sum(float v, float* s) {
    int t = threadIdx.x;
    s[t] = v; __syncthreads();
    for (int off = blockDim.x >> 1; off > 0; off >>= 1) {
        if (t < off) s[t] += s[t + off];
        __syncthreads();
    }
    float r = s[0]; __syncthreads();
    return r;
}
__device__ __forceinline__ float block_reduce_max(float v, float* s) {
    int t = threadIdx.x;
    s[t] = v; __syncthreads();
    for (int off = blockDim.x >> 1; off > 0; off >>= 1) {
        if (t < off) s[t] = fmaxf(s[t], s[t + off]);
        __syncthreads();
    }
    float r = s[0]; __syncthreads();
    return r;
}
__device__ __forceinline__ float block_reduce_min(float v, float* s) {
    int t = threadIdx.x;
    s[t] = v; __syncthreads();
    for (int off = blockDim.x >> 1; off > 0; off >>= 1) {
        if (t < off) s[t] = fminf(s[t], s[t + off]);
        __syncthreads();
    }
    float r = s[0]; __syncthreads();
    return r;
}

// ---------------------------------------------------------------------------
// Kernel 1: column dots  dot[j] = sum_i k[i]*MK[i][j]  via V_WMMA_F32_16X16X4_F32
//           and column sum-of-squares nsq[j] = sum_i MK[i][j]^2.
// Grid: NDIM/16 blocks (one per 16-column tile), 256 threads (8 waves).
// Each wave covers NDIM/8 = 1024 rows (256 WMMA steps of K=4); MK read once,
// coalesced along rows. A = k broadcast across all 16 M rows, so every row of
// the 16x16 accumulator holds the same 16 column dots.
// ---------------------------------------------------------------------------
__global__ __launch_bounds__(256)
void sim_dot_kernel(const float* __restrict__ MK,
                    const float* __restrict__ kvec,
                    float* __restrict__ ws_dot,
                    float* __restrict__ ws_nsq) {
    const int j0   = blockIdx.x * 16;
    const int wave = threadIdx.x >> 5;
    const int lane = threadIdx.x & 31;
    const int half = lane >> 4;        // 0: K=0,1   1: K=2,3   (16x4 f32 A/B layout)
    const int l16  = lane & 15;
    const int r0   = half * 2;
    const int row_base = wave * (NDIM / 8);   // 1024 rows per wave

    v8f  c   = {};     // f32 16x16 accumulator (8 VGPRs)
    float nsq = 0.f;

    for (int it = 0; it < (NDIM / 8) / 4; ++it) {
        const int k0 = row_base + it * 4;
        v2f a, b;
        // A (16x4, rows broadcast): this lane-half supplies K = r0, r0+1
        a.x = kvec[k0 + r0];
        a.y = kvec[k0 + r0 + 1];
        // B (4x16): element [K][N] with N = l16
        b.x = MK[(size_t)(k0 + r0) * NDIM + j0 + l16];
        b.y = MK[(size_t)(k0 + r0 + 1) * NDIM + j0 + l16];
        nsq = fmaf(b.x, b.x, nsq);
        nsq = fmaf(b.y, b.y, nsq);
        c = __builtin_amdgcn_wmma_f32_16x16x4_f32(
                /*neg_a=*/false, a, /*neg_b=*/false, b,
                /*c_mod=*/(short)0, c, /*reuse_a=*/false, /*reuse_b=*/false);
    }

    // Column sum-of-squares: lane l (K=0,1 partial) + lane l+16 (K=2,3 partial)
    nsq += __shfl_xor(nsq, 16, 32);

    __shared__ float sdot[8][16];
    __shared__ float snsq[8][16];
    if (lane < 16) {
        sdot[wave][l16] = c[0];   // C VGPR0, lanes 0-15 => M=0, N=lane
        snsq[wave][l16] = nsq;
    }
    __syncthreads();
    if (threadIdx.x < 16) {
        float d = 0.f, n = 0.f;
        for (int w = 0; w < 8; ++w) { d += sdot[w][threadIdx.x]; n += snsq[w][threadIdx.x]; }
        ws_dot[j0 + threadIdx.x] = d;
        ws_nsq[j0 + threadIdx.x] = n;
    }
}

// ---------------------------------------------------------------------------
// Kernel 2: all O(D) work in one 1024-thread block (8 elems/thread, strided).
// Computes: ||k||, sim, softmax -> w_r, argmin(w_u) (first occurrence),
// w_u_new, beta, w_w_new, min(w_u_new) -> w_lu_new (strictly-less => zeros),
// and broadcast row-vectors addmk = w_u_new*k, addmu = w_w_new*u.
// ---------------------------------------------------------------------------
__global__ __launch_bounds__(1024)
void small_vec_kernel(const float* __restrict__ kvec,
                      const float* __restrict__ uvec,
                      const float* __restrict__ w_w,
                      const float* __restrict__ w_u,
                      const float* __restrict__ w_lu,
                      const float* __restrict__ beta_param,
                      const float* __restrict__ ws_dot,
                      const float* __restrict__ ws_nsq,
                      float* __restrict__ ws_wr,
                      float* __restrict__ ws_addmk,
                      float* __restrict__ ws_addmu,
                      int*   __restrict__ ws_minidx,
                      float* __restrict__ out_ww,
                      float* __restrict__ out_wu,
                      float* __restrict__ out_wlu) {
    const int t = threadIdx.x;
    const int R = NDIM / 1024;   // 8
    __shared__ float sred[1024];
    __shared__ int   sidx[1024];

    // ||k||
    float kv[8];
    float ks = 0.f;
    for (int r = 0; r < R; ++r) {
        int j = t + r * 1024;
        kv[r] = kvec[j];
        ks = fmaf(kv[r], kv[r], ks);
    }
    const float knorm = sqrtf(block_reduce_sum(ks, sred));

    // sim + stable softmax
    float sim[8];
    float mx = -__builtin_inff();
    for (int r = 0; r < R; ++r) {
        int j = t + r * 1024;
        sim[r] = ws_dot[j] / (knorm * sqrtf(ws_nsq[j]));
        mx = fmaxf(mx, sim[r]);
    }
    mx = block_reduce_max(mx, sred);
    float e[8];
    float es = 0.f;
    for (int r = 0; r < R; ++r) { e[r] = expf(sim[r] - mx); es += e[r]; }
    es = block_reduce_sum(es, sred);
    float wr[8];
    for (int r = 0; r < R; ++r) {
        int j = t + r * 1024;
        wr[r] = e[r] / es;
        ws_wr[j] = wr[r];
    }

    // argmin(w_u) with first-occurrence tie-break (jnp.argmin semantics)
    float wu[8];
    float mnv = __builtin_inff();
    int   mni = NDIM;
    for (int r = 0; r < R; ++r) {
        int j = t + r * 1024;
        wu[r] = w_u[j];
        if (wu[r] < mnv || (wu[r] == mnv && j < mni)) { mnv = wu[r]; mni = j; }
    }
    sred[t] = mnv; sidx[t] = mni;
    __syncthreads();
    for (int off = 512; off > 0; off >>= 1) {
        if (t < off) {
            float vo = sred[t + off]; int io = sidx[t + off];
            if (vo < sred[t] || (vo == sred[t] && io < sidx[t])) { sred[t] = vo; sidx[t] = io; }
        }
        __syncthreads();
    }
    const int min_idx = sidx[0];
    __syncthreads();
    if (t == 0) *ws_minidx = min_idx;

    // w_u_new and its min
    float wun[8];
    float mn2 = __builtin_inff();
    for (int r = 0; r < R; ++r) {
        int j = t + r * 1024;
        wun[r] = GAMMA_F * wu[r] + wr[r] + w_w[j];
        out_wu[j] = wun[r];
        mn2 = fminf(mn2, wun[r]);
    }
    mn2 = block_reduce_min(mn2, sred);

    const float beta = 1.f / (1.f + expf(-beta_param[0]));
    for (int r = 0; r < R; ++r) {
        int j = t + r * 1024;
        float wwn = beta * wr[r] + (1.f - beta) * w_lu[j];
        out_ww[j]  = wwn;
        out_wlu[j] = (wun[r] < mn2) ? 1.f : 0.f;   // strictly-less, faithful
        ws_addmk[j] = wun[r] * kv[r];
        ws_addmu[j] = wwn * uvec[j];
    }
}

// ---------------------------------------------------------------------------
// Kernel 3: fused streaming pass, one block per row (8192 blocks x 256 thr):
//   mkv = MK[i][j] (col min_idx zeroed)
//   u_final[i] += mkv * w_r[j]            (read BEFORE write, as reference)
//   mk_new[i][j] = mkv + addmk[j]
//   mu_new[i][j] = (MU[i][j] col-zeroed) + addmu[j]
// float4 vectorized; w_r/addmk/addmu (96 KB) stay hot in L2.
// ---------------------------------------------------------------------------
__global__ __launch_bounds__(256)
void stream_update_kernel(const float* __restrict__ MK,
                          const float* __restrict__ MU,
                          const float* __restrict__ ws_wr,
                          const float* __restrict__ ws_addmk,
                          const float* __restrict__ ws_addmu,
                          const int*   __restrict__ ws_minidx,
                          float* __restrict__ out_ufinal,
                          float* __restrict__ out_mk,
                          float* __restrict__ out_mu) {
    const int i  = blockIdx.x;
    const int t  = threadIdx.x;
    const int mi = *ws_minidx;
    const int mi4 = mi >> 2, mic = mi & 3;
    const size_t rb = (size_t)i * NDIM;

    const float4* mk4  = (const float4*)(MK + rb);
    const float4* mu4  = (const float4*)(MU + rb);
    float4*       omk4 = (float4*)(out_mk + rb);
    float4*       omu4 = (float4*)(out_mu + rb);
    const float4* wr4  = (const float4*)ws_wr;
    const float4* amk4 = (const float4*)ws_addmk;
    const float4* amu4 = (const float4*)ws_addmu;

    float acc = 0.f;
    for (int it = 0; it < NDIM / 4 / 256; ++it) {   // 8 iterations
        const int j4 = t + it * 256;

        float4 m = mk4[j4];
        if (j4 == mi4) ((float*)&m)[mic] = 0.f;
        const float4 w = wr4[j4];
        acc = fmaf(m.x, w.x, acc);
        acc = fmaf(m.y, w.y, acc);
        acc = fmaf(m.z, w.z, acc);
        acc = fmaf(m.w, w.w, acc);
        const float4 a = amk4[j4];
        float4 o;
        o.x = m.x + a.x; o.y = m.y + a.y; o.z = m.z + a.z; o.w = m.w + a.w;
        omk4[j4] = o;

        float4 q = mu4[j4];
        if (j4 == mi4) ((float*)&q)[mic] = 0.f;
        const float4 b = amu4[j4];
        float4 p;
        p.x = q.x + b.x; p.y = q.y + b.y; p.z = q.z + b.z; p.w = q.w + b.w;
        omu4[j4] = p;
    }

    __shared__ float s[256];
    s[t] = acc; __syncthreads();
    for (int off = 128; off > 0; off >>= 1) {
        if (t < off) s[t] += s[t + off];
        __syncthreads();
    }
    if (t == 0) out_ufinal[i] = s[0];
}

// ---------------------------------------------------------------------------
// Launch
// ---------------------------------------------------------------------------
extern "C" void kernel_launch(void* const* d_in, const int* in_sizes, int n_in,
                              void* d_out, int out_size, void* d_ws, size_t ws_size,
                              hipStream_t stream) {
    const float* kvec  = (const float*)d_in[0];
    const float* uvec  = (const float*)d_in[1];
    const float* MK    = (const float*)d_in[2];
    const float* MU    = (const float*)d_in[3];
    const float* w_w   = (const float*)d_in[4];
    const float* w_u   = (const float*)d_in[5];
    const float* w_lu  = (const float*)d_in[6];
    const float* betap = (const float*)d_in[7];

    float* out = (float*)d_out;
    const size_t DSQ = (size_t)NDIM * NDIM;
    float* out_ufinal = out;                       // (D,)
    float* out_mk     = out + NDIM;                // (D,D)
    float* out_mu     = out + NDIM + DSQ;          // (D,D)
    float* out_ww     = out + NDIM + 2 * DSQ;      // (D,)
    float* out_wu     = out_ww + NDIM;             // (D,)
    float* out_wlu    = out_wu + NDIM;             // (D,)

    float* ws       = (float*)d_ws;
    float* ws_dot   = ws;                          // (D,)
    float* ws_nsq   = ws + 1 * NDIM;               // (D,)
    float* ws_wr    = ws + 2 * NDIM;               // (D,)
    float* ws_addmk = ws + 3 * NDIM;               // (D,)
    float* ws_addmu = ws + 4 * NDIM;               // (D,)
    int*   ws_mi    = (int*)(ws + 5 * NDIM);       // 1 int

    sim_dot_kernel<<<NDIM / 16, 256, 0, stream>>>(MK, kvec, ws_dot, ws_nsq);
    small_vec_kernel<<<1, 1024, 0, stream>>>(kvec, uvec, w_w, w_u, w_lu, betap,
                                             ws_dot, ws_nsq, ws_wr, ws_addmk,
                                             ws_addmu, ws_mi,
                                             out_ww, out_wu, out_wlu);
    stream_update_kernel<<<NDIM, 256, 0, stream>>>(MK, MU, ws_wr, ws_addmk, ws_addmu,
                                                   ws_mi, out_ufinal, out_mk, out_mu);
}